// _volume_render_torch_88450556494131
// MI455X (gfx1250) — compile-verified
//
#include <hip/hip_runtime.h>
#include <hip/hip_bf16.h>

// CDNA5 / gfx1250 volume rendering:
//   one wave32 per ray; wave-scan for transmittance;
//   feats streamed HBM -> LDS via GLOBAL_LOAD_ASYNC_TO_LDS_B128 (double-buffered,
//   ASYNCcnt-tracked); einsum via V_WMMA_F32_16X16X4_F32 fed from LDS.

typedef float v2f __attribute__((ext_vector_type(2)));
typedef float v8f __attribute__((ext_vector_type(8)));

static constexpr int kS  = 128;  // samples per ray
static constexpr int kF  = 32;   // features
static constexpr int kTS = 16;   // samples per LDS tile  (tile = kTS*kF*4 = 2048 B)
static constexpr int kNT = kS / kTS;  // 8 tiles per ray

__global__ __launch_bounds__(32) void volrend_wmma_kernel(
    const float* __restrict__ depths,        // [nray, kS]
    const float* __restrict__ sigma,         // [nray, kS]
    const float* __restrict__ feats,         // [nray, kS, kF]
    const unsigned char* __restrict__ valid, // [nray] (bool, 1 byte)
    float* __restrict__ out_feats,           // [nray, kF]
    float* __restrict__ out_depth,           // [nray]
    float* __restrict__ out_ws,              // [nray]
    float* __restrict__ out_alpha,           // [nray]
    int nray)
{
    const int ray = blockIdx.x;
    if (ray >= nray) return;
    const int lane = threadIdx.x;            // 0..31, wave32

    __shared__ float w_lds[kS];              // weights, 512 B
    __shared__ float fbuf[2][kTS * kF];      // double-buffered feats tiles, 2x2 KB

    const float* __restrict__ fray = feats + (size_t)ray * kS * kF;

    // LDS byte offsets (generic LDS pointer: low 32 bits are the LDS offset)
    const unsigned ldsA = (unsigned)(uintptr_t)(&fbuf[0][0]) + (unsigned)lane * 16u;
    const unsigned ldsB = (unsigned)(uintptr_t)(&fbuf[1][0]) + (unsigned)lane * 16u;

    // Async-copy one 2 KB tile (kTS x kF f32) into an LDS buffer.
    // 32 lanes x b128 = 512 B per instruction; offset: advances BOTH global
    // and LDS addresses (CDNA5 ISA), so 4 instructions/tile.
    auto issue_tile = [&](int tile, unsigned lds_addr) {
        const float* g = fray + (size_t)tile * kTS * kF + lane * 4;
        asm volatile(
            "global_load_async_to_lds_b128 %0, %1, off\n\t"
            "global_load_async_to_lds_b128 %0, %1, off offset:512\n\t"
            "global_load_async_to_lds_b128 %0, %1, off offset:1024\n\t"
            "global_load_async_to_lds_b128 %0, %1, off offset:1536"
            :: "v"(lds_addr), "v"(g) : "memory");
    };

    // Prime the pipeline: tiles 0 and 1 in flight while we do the scan phase.
    issue_tile(0, ldsA);
    issue_tile(1, ldsB);

    // ---------------- Phase 1: free energy, scan, weights ----------------
    // Lane L owns samples s = 4L .. 4L+3 (contiguous float4 loads, coalesced).
    const size_t rbase = (size_t)ray * kS;
    const float4 dd = *(const float4*)(depths + rbase + 4 * lane);
    const float4 sg = *(const float4*)(sigma  + rbase + 4 * lane);

    // depth[4L+4] comes from the next lane's first element (unused on lane 31).
    const float nd = __shfl_down(dd.x, 1, 32);

    const float fe0 = (dd.y - dd.x) * sg.x;
    const float fe1 = (dd.z - dd.y) * sg.y;
    const float fe2 = (dd.w - dd.z) * sg.z;
    const float fe3 = (lane == 31) ? 0.0f : (nd - dd.w) * sg.w;  // last dist = 0

    // lane-local inclusive prefix of free energy
    const float p0 = fe0;
    const float p1 = p0 + fe1;
    const float p2 = p1 + fe2;
    const float p3 = p2 + fe3;

    // wave32 inclusive scan of lane totals
    float t = p3;
    #pragma unroll
    for (int off = 1; off < 32; off <<= 1) {
        const float y = __shfl_up(t, off, 32);
        if (lane >= off) t += y;
    }
    const float base_excl = t - p3;   // sum of fe over all preceding lanes

    // exclusive cumsum at each owned sample
    const float e0 = base_excl;
    const float e1 = base_excl + p0;
    const float e2 = base_excl + p1;
    const float e3 = base_excl + p2;

    // alpha = 1 - exp(-fe); T = exp(-excl_cumsum); w = alpha * T
    const float a0 = 1.0f - __expf(-fe0);
    const float a1 = 1.0f - __expf(-fe1);
    const float a2 = 1.0f - __expf(-fe2);
    const float a3 = 1.0f - __expf(-fe3);

    const float w0 = a0 * __expf(-e0);
    const float w1 = a1 * __expf(-e1);
    const float w2 = a2 * __expf(-e2);
    const float w3 = a3 * __expf(-e3);

    w_lds[4 * lane + 0] = w0;
    w_lds[4 * lane + 1] = w1;
    w_lds[4 * lane + 2] = w2;
    w_lds[4 * lane + 3] = w3;

    // per-lane scalar partials
    float wsum = w0 + w1 + w2 + w3;
    float dsum = w0 * dd.x + w1 * dd.y + w2 * dd.z + w3 * dd.w;
    float asum = w0 * a0 + w1 * a1 + w2 * a2 + w3 * a3;

    __syncthreads();   // single-wave WG: dscnt wait (barrier degenerates to NOP)

    // butterfly reduce across the wave
    #pragma unroll
    for (int off = 16; off >= 1; off >>= 1) {
        wsum += __shfl_xor(wsum, off, 32);
        dsum += __shfl_xor(dsum, off, 32);
        asum += __shfl_xor(asum, off, 32);
    }

    // ---------------- Phase 2: einsum via V_WMMA_F32_16X16X4_F32 ----------------
    // D = A(16x4) * B(4x16) + C, f32 everywhere.
    //   A[m,k] = w[c+k]   (replicated over m -> every D row equals out[f])
    //   B[k,n] = feats[c+k, fbase+n]   (fed from the LDS tile)
    // f32 A 16x4 layout : lane m (<16): VGPR0=A[m,0], VGPR1=A[m,1]; lane m+16: K=2,3
    // f32 B 4x16 layout : lanes 0-15 rows K=0/1, lanes 16-31 rows K=2/3, col = lane&15
    const int ko   = (lane < 16) ? 0 : 2;
    const int fcol = lane & 15;

    v8f acc0 = {};   // features 0..15
    v8f acc1 = {};   // features 16..31

    for (int tile = 0; tile < kNT; ++tile) {
        // Retire exactly this tile's 4 async loads; keep the next tile in flight.
        if (tile < kNT - 1)
            asm volatile("s_wait_asynccnt 0x4" ::: "memory");
        else
            asm volatile("s_wait_asynccnt 0x0" ::: "memory");

        const float* __restrict__ lb = (tile & 1) ? &fbuf[1][0] : &fbuf[0][0];

        #pragma unroll
        for (int cl = 0; cl < kTS; cl += 4) {
            const int c = tile * kTS + cl;
            v2f A;
            A[0] = w_lds[c + ko];
            A[1] = w_lds[c + ko + 1];

            const float* fb = lb + (cl + ko) * kF + fcol;
            v2f B0, B1;
            B0[0] = fb[0];          // feats[c+ko,   fcol]
            B0[1] = fb[kF];         // feats[c+ko+1, fcol]
            B1[0] = fb[16];        // feats[c+ko,   16+fcol]
            B1[1] = fb[kF + 16];   // feats[c+ko+1, 16+fcol]

            acc0 = __builtin_amdgcn_wmma_f32_16x16x4_f32(
                false, A, false, B0, (short)0, acc0, false, false);
            acc1 = __builtin_amdgcn_wmma_f32_16x16x4_f32(
                false, A, false, B1, (short)0, acc1, false, false);
        }

        // Refill the buffer we just drained with tile+2 (WAR-safe: the ds reads
        // above are waited before the WMMAs consume them, and the volatile asm
        // with a memory clobber cannot be hoisted across them).
        if (tile + 2 < kNT)
            issue_tile(tile + 2, (tile & 1) ? ldsB : ldsA);
    }

    // ---------------- Outputs ----------------
    const float m = valid[ray] ? 1.0f : 0.0f;

    // C/D layout: VGPR0 lanes 0-15 = row M=0 (N=lane); lanes 16-31 = row M=8
    // (identical by replication). lane<16 -> out[f=lane] from acc0,
    // lane>=16 -> out[f=lane] from acc1 (f = 16 + (lane-16)).
    const float fv = (lane < 16) ? acc0[0] : acc1[0];
    out_feats[(size_t)ray * kF + lane] = fv * m;

    if (lane == 0) {
        out_depth[ray] = dsum * m;
        out_ws[ray]    = wsum * m;
        out_alpha[ray] = asum * m;
    }
}

extern "C" void kernel_launch(void* const* d_in, const int* in_sizes, int n_in,
                              void* d_out, int out_size, void* d_ws, size_t ws_size,
                              hipStream_t stream) {
    const float* depths = (const float*)d_in[0];          // [B,R,S]
    const float* sigma  = (const float*)d_in[1];          // [B,R,S]
    const float* feats  = (const float*)d_in[2];          // [B,R,S,F]
    // d_in[3] = max_depths: unused by the reference math
    const unsigned char* valid = (const unsigned char*)d_in[4];  // [B,R] bool

    const int nray = in_sizes[0] / kS;                    // B*R = 8192

    float* out       = (float*)d_out;                     // concatenated outputs
    float* out_feats = out;                               // [nray, F]
    float* out_depth = out + (size_t)nray * kF;           // [nray]
    float* out_ws    = out_depth + nray;                  // [nray]
    float* out_alpha = out_ws + nray;                     // [nray]

    volrend_wmma_kernel<<<nray, 32, 0, stream>>>(
        depths, sigma, feats, valid,
        out_feats, out_depth, out_ws, out_alpha, nray);
}